// Custom_GIN_21947282883021
// MI455X (gfx1250) — compile-verified
//
#include <hip/hip_runtime.h>

// GIN forward, MI455X / gfx1250, wave32.
//   N_NODES=10000, N_EDGES=160000, D=512 (=32*16), 3 layers.
// Per layer: agg = x + scatter_sum(x[src] -> dst); h = relu(agg@W0+b0)@W1+b1;
// relu between layers. GEMMs run on V_WMMA_F32_16X16X4_F32 (native fp32
// matrix path, matches the fp32 reference precision).
//
// GEMM tiling: block = 8 waves handles a 32-row M panel (M padded to 10016 =
// 313*32 in workspace). Each wave: 2 m-tiles x 4 n-tiles = 8 independent
// accumulator chains; B fragments (one global_load_b64 each, from K-pair-major
// packed weights) are reused across both m-tiles -> 0.5 global loads per WMMA.
// A panel staged in LDS in two K-halves (32x256, stride 260 floats ->
// conflict-free b64 fragment reads, 33.3KB static LDS).

#define NN      10000
#define NP      10016   // padded rows: 313 * 32
#define NE      160000
#define DD      512
#define NLAYERS 3

typedef __attribute__((ext_vector_type(2))) float v2f;
typedef __attribute__((ext_vector_type(8))) float v8f;

// ---------------------------------------------------------------- copy: A = h
__global__ __launch_bounds__(256) void gin_copy_f4(const float4* __restrict__ src,
                                                   float4* __restrict__ dst, int n4) {
    int i = blockIdx.x * blockDim.x + threadIdx.x;
    if (i < n4) dst[i] = src[i];
}

// ------------------------------------------------- scatter: A[dst] += h[src]
// 128 consecutive threads cover one edge's 512 floats (float4 per thread):
// the gather of h[src] is a coalesced 2KB row read; sums use f32 atomics in L2.
__global__ __launch_bounds__(256) void gin_scatter_add(const float* __restrict__ h,
                                                       const int* __restrict__ src,
                                                       const int* __restrict__ dst,
                                                       float* __restrict__ agg) {
    int t = blockIdx.x * blockDim.x + threadIdx.x;
    int e = t >> 7;            // 128 threads per edge
    if (e >= NE) return;
    int c = (t & 127) << 2;    // 4 floats per thread
    int s = src[e];
    int d = dst[e];
    const float4 v = *(const float4*)(h + (size_t)s * DD + c);
    float* o = agg + (size_t)d * DD + c;
    atomicAdd(o + 0, v.x);
    atomicAdd(o + 1, v.y);
    atomicAdd(o + 2, v.z);
    atomicAdd(o + 3, v.w);
}

// --------------------------------------- weight repack: K-pair-major for WMMA
// Wt (float2)[p * 512 + c] = ( W[2p][c], W[2p+1][c] ),  p in [0,256), c in [0,512)
__global__ __launch_bounds__(256) void gin_pack_w(const float* __restrict__ W,
                                                  float2* __restrict__ Wt) {
    int t = blockIdx.x * blockDim.x + threadIdx.x;  // 0 .. 256*512-1
    int p = t >> 9;
    int c = t & 511;
    float2 v;
    v.x = W[(size_t)(2 * p) * DD + c];
    v.y = W[(size_t)(2 * p + 1) * DD + c];
    Wt[(size_t)p * DD + c] = v;
}

// ------------------------------------------------------------- WMMA fp32 GEMM
// out(rows x DD) = act( A(rows x DD) @ W + bias ), W given packed (Wt).
// grid.x = NP/32 = 313. Stores guarded to rows < NN.
#define LDSS 260   // 256 + 4 floats pad; bank(m) = (4m + k) mod 64, conflict-free

__global__ __launch_bounds__(256) void gin_gemm_wmma(const float* __restrict__ A,
                                                     const float* __restrict__ Wt,
                                                     const float* __restrict__ bias,
                                                     float* __restrict__ out,
                                                     int apply_relu) {
    __shared__ float lds[32 * LDSS];   // 33,280 B

    const int mbase = blockIdx.x * 32;
    const int tid   = threadIdx.x;     // 0..255
    const int wave  = tid >> 5;        // 0..7
    const int lane  = tid & 31;        // wave32

    // ISA fragment layout (wave32):
    //   A 16x4 f32: lane m = lane&15; lane-half selects K pair {0,1} vs {2,3}
    //   B 4x16 f32: same K mapping, N = lane&15
    //   C/D 16x16:  VGPR v -> row = v + 8*(lane>=16), col = lane&15
    const int m        = lane & 15;
    const int laneHalf = lane >> 4;        // 0 or 1
    const int khalf    = laneHalf << 1;    // 0 or 2
    const int n0       = wave * 64;        // this wave's 64-column slice
    const int col      = n0 + m;

    v8f c00 = {}, c01 = {}, c02 = {}, c03 = {};   // m-tile 0, n-tiles 0..3
    v8f c10 = {}, c11 = {}, c12 = {}, c13 = {};   // m-tile 1, n-tiles 0..3

    for (int kc = 0; kc < 2; ++kc) {
        // ---- stage A K-half: 32 rows x 256 cols, coalesced float4 loads
        __syncthreads();   // protect previous half's reads
        {
            const float* Ag = A + (size_t)mbase * DD + kc * 256;
#pragma unroll
            for (int i = 0; i < 8; ++i) {
                int e   = i * 256 + tid;   // float4 index (2048 per half)
                int row = e >> 6;          // 64 float4 per row
                int c4  = e & 63;
                *(float4*)&lds[row * LDSS + c4 * 4] =
                    *(const float4*)(Ag + (size_t)row * DD + c4 * 4);
            }
        }
        __syncthreads();

        for (int kk = 0; kk < 256; kk += 4) {
            const int k0 = kc * 256 + kk;  // global K for B

            v2f a0, a1;
            const float* ap0 = &lds[m * LDSS + kk + khalf];
            const float* ap1 = &lds[(16 + m) * LDSS + kk + khalf];
            a0.x = ap0[0]; a0.y = ap0[1];
            a1.x = ap1[0]; a1.y = ap1[1];

            // packed B: pair p = k0/2 + laneHalf; one b64 per fragment,
            // reused by both m-tiles
            const float* bp = Wt + ((size_t)((k0 >> 1) + laneHalf) * DD + col) * 2;
            v2f b0, b1, b2, b3;
            b0.x = bp[0];  b0.y = bp[1];
            b1.x = bp[32]; b1.y = bp[33];
            b2.x = bp[64]; b2.y = bp[65];
            b3.x = bp[96]; b3.y = bp[97];

            c00 = __builtin_amdgcn_wmma_f32_16x16x4_f32(false, a0, false, b0, (short)0, c00, false, false);
            c01 = __builtin_amdgcn_wmma_f32_16x16x4_f32(false, a0, false, b1, (short)0, c01, false, false);
            c02 = __builtin_amdgcn_wmma_f32_16x16x4_f32(false, a0, false, b2, (short)0, c02, false, false);
            c03 = __builtin_amdgcn_wmma_f32_16x16x4_f32(false, a0, false, b3, (short)0, c03, false, false);
            c10 = __builtin_amdgcn_wmma_f32_16x16x4_f32(false, a1, false, b0, (short)0, c10, false, false);
            c11 = __builtin_amdgcn_wmma_f32_16x16x4_f32(false, a1, false, b1, (short)0, c11, false, false);
            c12 = __builtin_amdgcn_wmma_f32_16x16x4_f32(false, a1, false, b2, (short)0, c12, false, false);
            c13 = __builtin_amdgcn_wmma_f32_16x16x4_f32(false, a1, false, b3, (short)0, c13, false, false);
        }
    }

    // ---- epilogue: bias (+ optional relu), stores guarded to real rows
    const float bb0 = bias[n0 + m];
    const float bb1 = bias[n0 + 16 + m];
    const float bb2 = bias[n0 + 32 + m];
    const float bb3 = bias[n0 + 48 + m];
    const int rb0 = mbase + (laneHalf << 3);        // m-tile 0 rows
    const int rb1 = mbase + 16 + (laneHalf << 3);   // m-tile 1 rows
#pragma unroll
    for (int v = 0; v < 8; ++v) {
        int r0 = rb0 + v;
        if (r0 < NN) {
            float t0 = c00[v] + bb0, t1 = c01[v] + bb1;
            float t2 = c02[v] + bb2, t3 = c03[v] + bb3;
            if (apply_relu) {
                t0 = fmaxf(t0, 0.0f); t1 = fmaxf(t1, 0.0f);
                t2 = fmaxf(t2, 0.0f); t3 = fmaxf(t3, 0.0f);
            }
            float* orow = out + (size_t)r0 * DD + n0 + m;
            orow[0] = t0; orow[16] = t1; orow[32] = t2; orow[48] = t3;
        }
        int r1 = rb1 + v;
        if (r1 < NN) {
            float t0 = c10[v] + bb0, t1 = c11[v] + bb1;
            float t2 = c12[v] + bb2, t3 = c13[v] + bb3;
            if (apply_relu) {
                t0 = fmaxf(t0, 0.0f); t1 = fmaxf(t1, 0.0f);
                t2 = fmaxf(t2, 0.0f); t3 = fmaxf(t3, 0.0f);
            }
            float* orow = out + (size_t)r1 * DD + n0 + m;
            orow[0] = t0; orow[16] = t1; orow[32] = t2; orow[48] = t3;
        }
    }
}

// ---------------------------------------------------------------------- host
extern "C" void kernel_launch(void* const* d_in, const int* in_sizes, int n_in,
                              void* d_out, int out_size, void* d_ws, size_t ws_size,
                              hipStream_t stream) {
    const float* x  = (const float*)d_in[0];           // (10000, 512) f32
    const int*   ei = (const int*)d_in[1];             // (2, 160000) int32 (JAX x64 off)
    const float* Ws = (const float*)d_in[2];           // (3, 2, 512, 512) f32
    const float* bs = (const float*)d_in[3];           // (3, 2, 512) f32
    float*       out = (float*)d_out;                  // (10000, 512) f32

    const int* src = ei;            // edge_index[0]
    const int* dst = ei + NE;       // edge_index[1]

    float* A  = (float*)d_ws;                    // agg buffer     (NP*DD f32)
    float* B  = A + (size_t)NP * DD;             // MLP hidden     (NP*DD f32)
    float* H  = B + (size_t)NP * DD;             // layer output   (NP*DD f32)
    float* WT = H + (size_t)NP * DD;             // packed weights (DD*DD f32)

    const int n4          = NN * DD / 4;         // only real rows copied
    const int copy_blocks = (n4 + 255) / 256;
    const int scat_blocks = (NE * 128) / 256;    // 80000
    const int pack_blocks = (DD / 2) * DD / 256; // 512
    const int gemm_blocks = NP / 32;             // 313

    const float* h = x;
    for (int l = 0; l < NLAYERS; ++l) {
        // agg = h  (identity term: (1+eps)*x with eps=0)
        gin_copy_f4<<<copy_blocks, 256, 0, stream>>>((const float4*)h, (float4*)A, n4);
        // agg += scatter_sum(h[src] -> dst)
        gin_scatter_add<<<scat_blocks, 256, 0, stream>>>(h, src, dst, A);

        const float* W0 = Ws + ((size_t)l * 2 + 0) * DD * DD;
        const float* b0 = bs + ((size_t)l * 2 + 0) * DD;
        const float* W1 = Ws + ((size_t)l * 2 + 1) * DD * DD;
        const float* b1 = bs + ((size_t)l * 2 + 1) * DD;

        // B = relu(A @ W0 + b0)
        gin_pack_w<<<pack_blocks, 256, 0, stream>>>(W0, (float2*)WT);
        gin_gemm_wmma<<<gemm_blocks, 256, 0, stream>>>(A, WT, b0, B, 1);

        // H (or out) = B @ W1 + b1, relu between layers
        gin_pack_w<<<pack_blocks, 256, 0, stream>>>(W1, (float2*)WT);
        float* o = (l == NLAYERS - 1) ? out : H;
        gin_gemm_wmma<<<gemm_blocks, 256, 0, stream>>>(B, WT, b1, o, (l < NLAYERS - 1) ? 1 : 0);

        h = H;
    }
}